// GAT_88029649699615
// MI455X (gfx1250) — compile-verified
//
#include <hip/hip_runtime.h>
#include <math.h>

typedef float v2f __attribute__((ext_vector_type(2)));
typedef float v8f __attribute__((ext_vector_type(8)));

#define ASTR 20  // padded LDS stride for A tile (conflict-free fragment reads)

// ---------------------------------------------------------------------------
// FP32 WMMA GEMM: C[M,N] = A[M,K] @ B[K,N], row-major, K % 16 == 0, N % BN == 0
// 256 threads = 8 waves, wave tile 32x32 (4 accumulators), block tile BMxBN.
// ---------------------------------------------------------------------------
template <int BM, int BN, int WROWS, int WCOLS>
__global__ __launch_bounds__(256) void gemm_wmma_f32(
    const float* __restrict__ A, const float* __restrict__ Bw,
    float* __restrict__ C, int M, int K, int N) {
  __shared__ float As[BM * ASTR];
  __shared__ float Bs[16 * BN];
  const int tid = threadIdx.x;
  const int lane = tid & 31;
  const int w = tid >> 5;
  const int wr = w / WCOLS;
  const int wc = w % WCOLS;
  const int lm = lane & 15;
  const int lh = lane >> 4;
  const int m0 = blockIdx.x * BM;
  const int n0 = blockIdx.y * BN;

  v8f acc[2][2] = {};

  for (int k0 = 0; k0 < K; k0 += 16) {
#pragma unroll
    for (int i = 0; i < BM / 64; ++i) {  // A tile: BM x 16 floats
      int f = tid + i * 256;             // float4 id
      int row = f >> 2;
      int cin = (f & 3) << 2;
      int gr = m0 + row;
      float4 v = make_float4(0.f, 0.f, 0.f, 0.f);
      if (gr < M) v = *(const float4*)&A[(size_t)gr * K + k0 + cin];
      *(float4*)&As[row * ASTR + cin] = v;
    }
#pragma unroll
    for (int i = 0; i < BN / 64; ++i) {  // B tile: 16 x BN floats
      int f = tid + i * 256;
      int row = f / (BN >> 2);
      int col = (f % (BN >> 2)) << 2;
      *(float4*)&Bs[row * BN + col] =
          *(const float4*)&Bw[(size_t)(k0 + row) * N + n0 + col];
    }
    __syncthreads();
#pragma unroll
    for (int kk = 0; kk < 16; kk += 4) {
      v2f afr[2], bfr[2];
#pragma unroll
      for (int mi = 0; mi < 2; ++mi) {
        int r = wr * 32 + mi * 16 + lm;
        afr[mi] = *(const v2f*)&As[r * ASTR + kk + 2 * lh];
      }
#pragma unroll
      for (int ni = 0; ni < 2; ++ni) {
        int c = wc * 32 + ni * 16 + lm;
        int rB = kk + 2 * lh;
        v2f b;
        b.x = Bs[rB * BN + c];
        b.y = Bs[(rB + 1) * BN + c];
        bfr[ni] = b;
      }
#pragma unroll
      for (int mi = 0; mi < 2; ++mi)
#pragma unroll
        for (int ni = 0; ni < 2; ++ni)
          acc[mi][ni] = __builtin_amdgcn_wmma_f32_16x16x4_f32(
              false, afr[mi], false, bfr[ni], (short)0, acc[mi][ni], false,
              false);
    }
    __syncthreads();
  }
#pragma unroll
  for (int mi = 0; mi < 2; ++mi)
#pragma unroll
    for (int ni = 0; ni < 2; ++ni) {
      int col = n0 + wc * 32 + ni * 16 + lm;
#pragma unroll
      for (int r = 0; r < 8; ++r) {
        int row = m0 + wr * 32 + mi * 16 + r + 8 * lh;
        if (row < M) C[(size_t)row * N + col] = acc[mi][ni][r];
      }
    }
}

// ---------------------------------------------------------------------------
// helpers
// ---------------------------------------------------------------------------
__device__ inline void edge_sd(const int* __restrict__ ei, int e, int E,
                               int& s, int& d) {
  if (e < E) {
    s = ei[e];
    d = ei[E + e];
  } else {
    s = d = e - E;  // self loops appended after real edges
  }
}

// monotone float <-> uint key for atomicMax on floats
__device__ inline unsigned fkey(float f) {
  int i = __float_as_int(f);
  return (unsigned)(i ^ ((i >> 31) | 0x80000000));
}
__device__ inline float funkey(unsigned k) {
  int i = (k & 0x80000000u) ? (int)(k ^ 0x80000000u) : ~(int)k;
  return __int_as_float(i);
}

// ---------------------------------------------------------------------------
// Layer-1 edge kernels (H=8, C=64, F=512)
// ---------------------------------------------------------------------------
__global__ __launch_bounds__(256) void e1_logit_k(
    const float* __restrict__ XL, const float* __restrict__ XR,
    const float* __restrict__ a1, const int* __restrict__ ei,
    float* __restrict__ L, unsigned* __restrict__ M1, int E, int NTOT) {
  int wid = (blockIdx.x * 256 + threadIdx.x) >> 5;  // wave per edge
  int lane = threadIdx.x & 31;
  int ET = E + NTOT;
  if (wid >= ET) return;
  int s, d;
  edge_sd(ei, wid, E, s, d);
  const float* xs = XL + (size_t)s * 512;
  const float* xd = XR + (size_t)d * 512;
#pragma unroll
  for (int h = 0; h < 8; ++h) {
    int c0 = h * 64 + lane;
    float t0 = xs[c0] + xd[c0];
    float t1 = xs[c0 + 32] + xd[c0 + 32];
    t0 = t0 > 0.f ? t0 : 0.2f * t0;  // leaky_relu 0.2
    t1 = t1 > 0.f ? t1 : 0.2f * t1;
    float p = t0 * a1[h * 64 + lane] + t1 * a1[h * 64 + lane + 32];
    for (int off = 16; off; off >>= 1) p += __shfl_xor(p, off, 32);
    if (lane == 0) {
      L[(size_t)wid * 8 + h] = p;
      atomicMax(&M1[(size_t)d * 8 + h], fkey(p));
    }
  }
}

__global__ __launch_bounds__(256) void e1_exp_k(
    float* __restrict__ L, const unsigned* __restrict__ M1,
    float* __restrict__ S, const int* __restrict__ ei, int E, int NTOT) {
  size_t i = (size_t)blockIdx.x * 256 + threadIdx.x;
  int ET = E + NTOT;
  if (i >= (size_t)ET * 8) return;
  int e = (int)(i >> 3), h = (int)(i & 7);
  int s, d;
  edge_sd(ei, e, E, s, d);
  (void)s;
  float p = expf(L[i] - funkey(M1[(size_t)d * 8 + h]));
  L[i] = p;  // store p over logit
  atomicAdd(&S[(size_t)d * 8 + h], p);
}

__global__ __launch_bounds__(256) void e1_agg_k(
    const float* __restrict__ XL, const float* __restrict__ P,
    const float* __restrict__ S, float* __restrict__ AGG,
    const int* __restrict__ ei, int E, int NTOT) {
  int wid = (blockIdx.x * 256 + threadIdx.x) >> 5;
  int lane = threadIdx.x & 31;
  int ET = E + NTOT;
  if (wid >= ET) return;
  int s, d;
  edge_sd(ei, wid, E, s, d);
  float alph[8];
#pragma unroll
  for (int h = 0; h < 8; ++h)
    alph[h] = P[(size_t)wid * 8 + h] / (S[(size_t)d * 8 + h] + 1e-16f);
  const float* xs = XL + (size_t)s * 512;
  float* ag = AGG + (size_t)d * 512;
#pragma unroll
  for (int i = 0; i < 16; ++i) {
    int c = lane + i * 32;
    atomicAdd(&ag[c], alph[c >> 6] * xs[c]);
  }
}

__global__ __launch_bounds__(256) void n1_act_k(float* __restrict__ AGG,
                                                const float* __restrict__ b1,
                                                int NTOT) {
  size_t i = (size_t)blockIdx.x * 256 + threadIdx.x;
  if (i >= (size_t)NTOT * 512) return;
  float v = AGG[i] + b1[i & 511];
  AGG[i] = v > 0.f ? v : (expf(v) - 1.f);  // elu
}

// ---------------------------------------------------------------------------
// Layer-2 edge kernels (H=1, C=64)
// ---------------------------------------------------------------------------
__global__ __launch_bounds__(256) void e2_logit_k(
    const float* __restrict__ XL, const float* __restrict__ XR,
    const float* __restrict__ a2, const int* __restrict__ ei,
    float* __restrict__ L, unsigned* __restrict__ M2, int E, int NTOT) {
  int wid = (blockIdx.x * 256 + threadIdx.x) >> 5;
  int lane = threadIdx.x & 31;
  int ET = E + NTOT;
  if (wid >= ET) return;
  int s, d;
  edge_sd(ei, wid, E, s, d);
  float t0 = XL[(size_t)s * 64 + lane] + XR[(size_t)d * 64 + lane];
  float t1 = XL[(size_t)s * 64 + lane + 32] + XR[(size_t)d * 64 + lane + 32];
  t0 = t0 > 0.f ? t0 : 0.2f * t0;
  t1 = t1 > 0.f ? t1 : 0.2f * t1;
  float p = t0 * a2[lane] + t1 * a2[lane + 32];
  for (int off = 16; off; off >>= 1) p += __shfl_xor(p, off, 32);
  if (lane == 0) {
    L[wid] = p;
    atomicMax(&M2[d], fkey(p));
  }
}

__global__ __launch_bounds__(256) void e2_exp_k(
    float* __restrict__ L, const unsigned* __restrict__ M2,
    float* __restrict__ S, const int* __restrict__ ei, int E, int NTOT) {
  int e = blockIdx.x * 256 + threadIdx.x;
  int ET = E + NTOT;
  if (e >= ET) return;
  int s, d;
  edge_sd(ei, e, E, s, d);
  (void)s;
  float p = expf(L[e] - funkey(M2[d]));
  L[e] = p;
  atomicAdd(&S[d], p);
}

__global__ __launch_bounds__(256) void e2_agg_k(
    const float* __restrict__ XL, const float* __restrict__ P,
    const float* __restrict__ S, float* __restrict__ AGG,
    const int* __restrict__ ei, int E, int NTOT) {
  size_t i = (size_t)blockIdx.x * 256 + threadIdx.x;
  int ET = E + NTOT;
  if (i >= (size_t)ET * 64) return;
  int e = (int)(i >> 6), c = (int)(i & 63);
  int s, d;
  edge_sd(ei, e, E, s, d);
  float alpha = P[e] / (S[d] + 1e-16f);
  atomicAdd(&AGG[(size_t)d * 64 + c], alpha * XL[(size_t)s * 64 + c]);
}

// selu + bias, and xw = h2 @ Wp (wave per node)
__global__ __launch_bounds__(256) void n2_act_k(float* __restrict__ AGG2,
                                                const float* __restrict__ b2,
                                                const float* __restrict__ Wp,
                                                float* __restrict__ XW,
                                                int NTOT) {
  int n = (blockIdx.x * 256 + threadIdx.x) >> 5;
  int lane = threadIdx.x & 31;
  if (n >= NTOT) return;
  const float SCL = 1.0507009873554805f, ALP = 1.6732632423543772f;
  float v0 = AGG2[(size_t)n * 64 + lane] + b2[lane];
  float v1 = AGG2[(size_t)n * 64 + lane + 32] + b2[lane + 32];
  v0 = SCL * (v0 > 0.f ? v0 : ALP * (expf(v0) - 1.f));
  v1 = SCL * (v1 > 0.f ? v1 : ALP * (expf(v1) - 1.f));
  AGG2[(size_t)n * 64 + lane] = v0;
  AGG2[(size_t)n * 64 + lane + 32] = v1;
  float pxw = v0 * Wp[lane] + v1 * Wp[lane + 32];
  for (int off = 16; off; off >>= 1) pxw += __shfl_xor(pxw, off, 32);
  if (lane == 0) XW[n] = pxw;
}

__global__ __launch_bounds__(256) void deg_k(float* __restrict__ DEG,
                                             const int* __restrict__ ei, int E,
                                             int NTOT) {
  int e = blockIdx.x * 256 + threadIdx.x;
  int ET = E + NTOT;
  if (e >= ET) return;
  int s, d;
  edge_sd(ei, e, E, s, d);
  (void)s;
  atomicAdd(&DEG[d], 1.f);
}

__global__ __launch_bounds__(256) void score_k(float* __restrict__ SCORE,
                                               const float* __restrict__ DEG,
                                               const float* __restrict__ XW,
                                               const int* __restrict__ ei,
                                               int E, int NTOT) {
  int e = blockIdx.x * 256 + threadIdx.x;
  int ET = E + NTOT;
  if (e >= ET) return;
  int s, d;
  edge_sd(ei, e, E, s, d);
  float ns = rsqrtf(fmaxf(DEG[s], 1.f));
  float nd = rsqrtf(fmaxf(DEG[d], 1.f));
  atomicAdd(&SCORE[d], ns * nd * XW[s]);
}

// ---------------------------------------------------------------------------
// Per-graph: top-K select (bitonic-sorted threshold), tanh-scaled max/mean
// pooling, linear + relu + log_softmax. One block per graph.
// ---------------------------------------------------------------------------
#define SORTN 8192
__global__ __launch_bounds__(256) void pool_k(
    const float* __restrict__ H2, const float* __restrict__ SCORE,
    const float* __restrict__ bp, const float* __restrict__ Wlin,
    const float* __restrict__ blin, float* __restrict__ out, int NPG, int KS) {
  __shared__ float sbuf[SORTN];
  __shared__ float redmax[256];
  __shared__ float redsum[256];
  __shared__ float pooled[128];
  __shared__ int counters[2];
  int g = blockIdx.x, tid = threadIdx.x;
  size_t base = (size_t)g * NPG;
  float bpv = bp[0];

  for (int i = tid; i < SORTN; i += 256)
    sbuf[i] = (i < NPG) ? (SCORE[base + i] + bpv) : -3.402823466e38f;
  if (tid < 2) counters[tid] = 0;
  __syncthreads();

  // bitonic sort, descending
  for (int k = 2; k <= SORTN; k <<= 1) {
    for (int j = k >> 1; j > 0; j >>= 1) {
      for (int i = tid; i < SORTN; i += 256) {
        int ixj = i ^ j;
        if (ixj > i) {
          float va = sbuf[i], vb = sbuf[ixj];
          bool desc = ((i & k) == 0);
          if (desc ? (va < vb) : (va > vb)) {
            sbuf[i] = vb;
            sbuf[ixj] = va;
          }
        }
      }
      __syncthreads();
    }
  }
  float thresh = sbuf[KS - 1];
  // count strictly greater than threshold
  int cg = 0;
  for (int i = tid; i < NPG; i += 256)
    if (sbuf[i] > thresh) cg++;
  atomicAdd(&counters[0], cg);
  __syncthreads();
  int quota = KS - counters[0];
  __syncthreads();
  // selection mask (reuse sbuf storage as ints)
  int* sel = (int*)sbuf;
  for (int i = tid; i < NPG; i += 256) {
    float sc = SCORE[base + i] + bpv;
    int take = 0;
    if (sc > thresh)
      take = 1;
    else if (sc == thresh)
      take = (atomicAdd(&counters[1], 1) < quota) ? 1 : 0;
    sel[i] = take;
  }
  __syncthreads();
  // pooling: 4 node-lanes x 64 dims
  int dd = tid & 63, nl = tid >> 6;
  float mx = -3.402823466e38f, sm = 0.f;
  for (int i = nl; i < NPG; i += 4) {
    if (sel[i]) {
      float t = tanhf(SCORE[base + i] + bpv);
      float v = H2[(base + i) * 64 + dd] * t;
      mx = fmaxf(mx, v);
      sm += v;
    }
  }
  redmax[tid] = mx;
  redsum[tid] = sm;
  __syncthreads();
  if (nl == 0) {
    float m = redmax[dd], s = redsum[dd];
#pragma unroll
    for (int q = 1; q < 4; ++q) {
      m = fmaxf(m, redmax[q * 64 + dd]);
      s += redsum[q * 64 + dd];
    }
    pooled[dd] = m;                    // gmp
    pooled[64 + dd] = s / (float)KS;   // gap
  }
  __syncthreads();
  if (tid == 0) {
    float l[3];
#pragma unroll
    for (int j = 0; j < 3; ++j) {
      float acc = blin[j];
      for (int c = 0; c < 128; ++c) acc += pooled[c] * Wlin[c * 3 + j];
      l[j] = acc > 0.f ? acc : 0.f;  // relu
    }
    float mm = fmaxf(l[0], fmaxf(l[1], l[2]));
    float lse =
        mm + logf(expf(l[0] - mm) + expf(l[1] - mm) + expf(l[2] - mm));
#pragma unroll
    for (int j = 0; j < 3; ++j) out[g * 3 + j] = l[j] - lse;
  }
}

// ---------------------------------------------------------------------------
extern "C" void kernel_launch(void* const* d_in, const int* in_sizes, int n_in,
                              void* d_out, int out_size, void* d_ws,
                              size_t ws_size, hipStream_t stream) {
  const float* x = (const float*)d_in[0];
  const float* Wl1 = (const float*)d_in[1];
  const float* Wr1 = (const float*)d_in[2];
  const float* a1 = (const float*)d_in[3];
  const float* b1 = (const float*)d_in[4];
  const float* Wl2 = (const float*)d_in[5];
  const float* Wr2 = (const float*)d_in[6];
  const float* a2 = (const float*)d_in[7];
  const float* b2 = (const float*)d_in[8];
  const float* Wp = (const float*)d_in[9];
  const float* bp = (const float*)d_in[10];
  const float* Wlin = (const float*)d_in[11];
  const float* blin = (const float*)d_in[12];
  const int* ei = (const int*)d_in[13];
  float* out = (float*)d_out;

  const int NTOT = in_sizes[14];      // 50000 nodes (batch array length)
  const int E = in_sizes[13] / 2;     // 400000 edges (self loops implicit)
  const int ET = E + NTOT;
  const int BG = 10;                  // graphs
  const int NPG = NTOT / BG;          // 5000 nodes per graph
  const int KS = (NPG + 1) / 2;       // top-k, ratio 0.5 (ceil)

  char* p = (char*)d_ws;
  size_t off = 0;
  auto take = [&](size_t bytes) -> char* {
    char* q = p + off;
    off += (bytes + 255) & ~(size_t)255;
    return q;
  };
  float* XL1 = (float*)take((size_t)NTOT * 512 * 4);
  char* R1 = take((size_t)NTOT * 512 * 4);  // XR1, recycled for layer 2
  float* XR1 = (float*)R1;
  float* AGG1 = (float*)take((size_t)NTOT * 512 * 4);  // -> h1 (elu)
  float* ELOG1 = (float*)take((size_t)ET * 8 * 4);
  unsigned* M1 = (unsigned*)take((size_t)NTOT * 8 * 4);
  float* S1 = (float*)take((size_t)NTOT * 8 * 4);
  float* ELOG2 = (float*)take((size_t)ET * 4);
  unsigned* M2 = (unsigned*)take((size_t)NTOT * 4);
  float* S2 = (float*)take((size_t)NTOT * 4);
  float* XW = (float*)take((size_t)NTOT * 4);
  float* DEG = (float*)take((size_t)NTOT * 4);
  float* SCORE = (float*)take((size_t)NTOT * 4);
  // layer-2 tensors overlaid on R1 (dead after e1_logit_k)
  float* XL2 = (float*)(R1);
  float* XR2 = (float*)(R1 + (size_t)NTOT * 64 * 4);
  float* AGG2 = (float*)(R1 + 2 * ((size_t)NTOT * 64 * 4));  // -> h2 (selu)

  hipMemsetAsync(M1, 0, (size_t)NTOT * 8 * 4, stream);
  hipMemsetAsync(S1, 0, (size_t)NTOT * 8 * 4, stream);
  hipMemsetAsync(AGG1, 0, (size_t)NTOT * 512 * 4, stream);
  hipMemsetAsync(M2, 0, (size_t)NTOT * 4, stream);
  hipMemsetAsync(S2, 0, (size_t)NTOT * 4, stream);
  hipMemsetAsync(DEG, 0, (size_t)NTOT * 4, stream);
  hipMemsetAsync(SCORE, 0, (size_t)NTOT * 4, stream);

  dim3 blk(256);
  const int ebw = (ET + 7) / 8;  // wave-per-edge grids

  // Layer 1 projections: [NTOT,1024] @ [1024,512]
  dim3 g1((NTOT + 63) / 64, 512 / 128);
  gemm_wmma_f32<64, 128, 2, 4><<<g1, blk, 0, stream>>>(x, Wl1, XL1, NTOT, 1024, 512);
  gemm_wmma_f32<64, 128, 2, 4><<<g1, blk, 0, stream>>>(x, Wr1, XR1, NTOT, 1024, 512);

  e1_logit_k<<<ebw, blk, 0, stream>>>(XL1, XR1, a1, ei, ELOG1, M1, E, NTOT);
  e1_exp_k<<<(int)(((size_t)ET * 8 + 255) / 256), blk, 0, stream>>>(ELOG1, M1, S1, ei, E, NTOT);
  e1_agg_k<<<ebw, blk, 0, stream>>>(XL1, ELOG1, S1, AGG1, ei, E, NTOT);
  n1_act_k<<<(int)(((size_t)NTOT * 512 + 255) / 256), blk, 0, stream>>>(AGG1, b1, NTOT);

  hipMemsetAsync(AGG2, 0, (size_t)NTOT * 64 * 4, stream);

  // Layer 2 projections: [NTOT,512] @ [512,64]
  dim3 g2((NTOT + 127) / 128, 1);
  gemm_wmma_f32<128, 64, 4, 2><<<g2, blk, 0, stream>>>(AGG1, Wl2, XL2, NTOT, 512, 64);
  gemm_wmma_f32<128, 64, 4, 2><<<g2, blk, 0, stream>>>(AGG1, Wr2, XR2, NTOT, 512, 64);

  e2_logit_k<<<ebw, blk, 0, stream>>>(XL2, XR2, a2, ei, ELOG2, M2, E, NTOT);
  e2_exp_k<<<(ET + 255) / 256, blk, 0, stream>>>(ELOG2, M2, S2, ei, E, NTOT);
  e2_agg_k<<<(int)(((size_t)ET * 64 + 255) / 256), blk, 0, stream>>>(XL2, ELOG2, S2, AGG2, ei, E, NTOT);
  n2_act_k<<<(NTOT + 7) / 8, blk, 0, stream>>>(AGG2, b2, Wp, XW, NTOT);

  deg_k<<<(ET + 255) / 256, blk, 0, stream>>>(DEG, ei, E, NTOT);
  score_k<<<(ET + 255) / 256, blk, 0, stream>>>(SCORE, DEG, XW, ei, E, NTOT);

  pool_k<<<BG, blk, 0, stream>>>(AGG2, SCORE, bp, Wlin, blin, out, NPG, KS);
}